// MixedEmbeddingV2_41429254537402
// MI455X (gfx1250) — compile-verified
//
#include <hip/hip_runtime.h>
#include <hip/hip_bf16.h>

// MixedEmbeddingV2: out[t, d] = emb[x[t], d] * col_scale[d]
// col_scale is piecewise-constant over 4 segments of 192 columns (suffix sums
// of the 4 mixture weights; dims = 192,384,576,768).
//
// Bandwidth-bound gather (~48 MB read + 48 MB write -> ~4.2 us @ 23.3 TB/s).
// CDNA5 path: issue all 8 token-row gathers per block up front with
// GLOBAL_LOAD_ASYNC_TO_LDS_B128 (ASYNCcnt-tracked, no VGPRs held across the
// gather latency), drain with s_wait_asynccnt N (compile-time immediates),
// scale, and write the write-once output with non-temporal 128-bit stores so
// it never evicts the L2-resident (154 MB < 192 MB) embedding table.

typedef float v4f __attribute__((ext_vector_type(4)));
typedef int   v4i __attribute__((ext_vector_type(4)));

static constexpr int D           = 768;          // MAX_EMBED_DIM
static constexpr int SEG         = 192;          // embed-dim step
static constexpr int VEC         = 4;            // float4 per thread
static constexpr int THREADS     = D / VEC;      // 192 threads = 6 wave32s
static constexpr int SEG_THREADS = SEG / VEC;    // 48 threads per segment
static constexpr int TPB         = 8;            // tokens per block

#if defined(__has_builtin)
#if __has_builtin(__builtin_amdgcn_global_load_async_to_lds_b128)
#define USE_ASYNC_GATHER 1
#endif
#endif
#ifndef USE_ASYNC_GATHER
#define USE_ASYNC_GATHER 0
#endif

#if USE_ASYNC_GATHER
// Builtin signature (from hipcc diagnostic): param0 = v4i __global* (AS1),
// param1 = v4i __shared* (AS3), then imm offset, imm cpol.
typedef __attribute__((address_space(1))) v4i gv4i;
typedef __attribute__((address_space(3))) v4i sv4i;

template <int N>
__device__ __forceinline__ void wait_asynccnt() {
#if __has_builtin(__builtin_amdgcn_s_wait_asynccnt)
    __builtin_amdgcn_s_wait_asynccnt(N);
#else
    asm volatile("s_wait_asynccnt %0" :: "i"(N) : "memory");
#endif
}

// Compile-time drain so the s_wait_asynccnt immediate is a true constant.
template <int I>
__device__ __forceinline__ void drain_from(const float* stage, float* out,
                                           int t0, int n_tokens, int tid, float scale)
{
    if constexpr (I < TPB) {
        wait_asynccnt<TPB - 1 - I>();          // tokens 0..I now resident in LDS
        const int t = t0 + I;
        if (t < n_tokens) {
            v4f v = *reinterpret_cast<const v4f*>(
                stage + (size_t)I * D + (size_t)tid * VEC);     // ds_load_b128
            v *= scale;
            v4f* dst = reinterpret_cast<v4f*>(out + (size_t)t * D) + tid;
            __builtin_nontemporal_store(v, dst);                // global_store_b128 th:NT
        }
        drain_from<I + 1>(stage, out, t0, n_tokens, tid, scale);
    }
}
#endif

__global__ __launch_bounds__(THREADS)
void mixed_embedding_gather_kernel(const int*   __restrict__ x,      // [n_tokens]
                                   const float* __restrict__ w,      // [4]
                                   const float* __restrict__ emb,    // [VOCAB, D]
                                   float*       __restrict__ out,    // [n_tokens, D]
                                   int n_tokens)
{
    // Suffix sums of the mixture weights -> per-segment scales (scalarized
    // to s_add_f32 by the compiler; w is block-uniform and L2-resident).
    const float w0 = w[0], w1 = w[1], w2 = w[2], w3 = w[3];
    const float s3 = w3;
    const float s2 = s3 + w2;
    const float s1 = s2 + w1;
    const float s0 = s1 + w0;

    const int tid = threadIdx.x;              // 0..191
    const int seg = tid / SEG_THREADS;        // 0..3 (float4 never straddles segs)
    const float scale = (seg == 0) ? s0 : (seg == 1) ? s1 : (seg == 2) ? s2 : s3;

    const int t0 = blockIdx.x * TPB;

    // Gather the 8 token row ids (block-uniform -> scalar loads).
    int rows[TPB];
    #pragma unroll
    for (int i = 0; i < TPB; ++i) {
        const int t = t0 + i;
        rows[i] = x[(t < n_tokens) ? t : (n_tokens - 1)];
    }

#if USE_ASYNC_GATHER
    // 24 KB staging buffer: 8 rows x 3 KB (320 KB LDS per WGP -> no pressure).
    __shared__ __align__(16) float stage[TPB][D];

    // Issue ALL eight row gathers immediately: 192 lanes x 16 B = one 3 KB row
    // per async op, ASYNCcnt-tracked, zero VGPRs held while in flight.
    #pragma unroll
    for (int i = 0; i < TPB; ++i) {
        const float* src = emb + (size_t)rows[i] * D + (size_t)tid * VEC;
        float*       dst = &stage[i][tid * VEC];
        __builtin_amdgcn_global_load_async_to_lds_b128(
            (gv4i*)src, (sv4i*)dst, /*offset=*/0, /*cpol=*/0);
    }

    // Drain in order. Each lane reads back exactly the 16 B it requested, so
    // per-wave ASYNCcnt ordering suffices -- no workgroup barrier needed.
    drain_from<0>(&stage[0][0], out, t0, n_tokens, tid, scale);
#else
    // Fallback: direct 128-bit gather with cross-token prefetch.
    #pragma unroll
    for (int i = 0; i < TPB; ++i) {
        const int t = t0 + i;
        if (t >= n_tokens) return;

        if (i + 1 < TPB && t + 1 < n_tokens) {
            __builtin_prefetch(emb + (size_t)rows[i + 1] * D + (size_t)tid * VEC, 0, 0);
        }

        const v4f* src = reinterpret_cast<const v4f*>(emb + (size_t)rows[i] * D) + tid;
        v4f v = *src;
        v *= scale;

        v4f* dst = reinterpret_cast<v4f*>(out + (size_t)t * D) + tid;
        __builtin_nontemporal_store(v, dst);
    }
#endif
}

extern "C" void kernel_launch(void* const* d_in, const int* in_sizes, int n_in,
                              void* d_out, int out_size, void* d_ws, size_t ws_size,
                              hipStream_t stream)
{
    (void)n_in; (void)out_size; (void)d_ws; (void)ws_size;

    const int*   x   = (const int*)  d_in[0];   // token ids [B*S]
    const float* w   = (const float*)d_in[1];   // mixture weights [4]
    const float* emb = (const float*)d_in[2];   // embedding table [VOCAB, 768]
    float*       out = (float*)      d_out;     // [B*S, 768]

    const int n_tokens = in_sizes[0];           // 8 * 2048 = 16384
    const int grid = (n_tokens + TPB - 1) / TPB;

    mixed_embedding_gather_kernel<<<grid, THREADS, 0, stream>>>(x, w, emb, out, n_tokens);
}